// BasicConvolutionBlock_25898652795311
// MI455X (gfx1250) — compile-verified
//
#include <hip/hip_runtime.h>

typedef float v2f __attribute__((ext_vector_type(2)));
typedef float v8f __attribute__((ext_vector_type(8)));

#define N_VOX   200000
#define N_PAIRS 100000
#define KOFF    27
#define INC     64
#define OUTC    64
#define BN_EPS  1e-5f
#define TILES   (N_PAIRS / 16)   // 6250 pair tiles of 16

__device__ __forceinline__ void atomicAddF32(float* p, float v) {
    // relaxed agent-scope fadd -> global_atomic_add_f32 on gfx1250
    __hip_atomic_fetch_add(p, v, __ATOMIC_RELAXED, __HIP_MEMORY_SCOPE_AGENT);
}

// ---------------------------------------------------------------------------
// Kernel 1: zero the conv accumulator (d_out) and the BN stats scratch (d_ws)
// ---------------------------------------------------------------------------
__global__ void zero_kernel(float* __restrict__ out, float* __restrict__ ws, int n4) {
    float4* o4 = (float4*)out;
    const float4 z = make_float4(0.f, 0.f, 0.f, 0.f);
    for (int i = blockIdx.x * blockDim.x + threadIdx.x; i < n4;
         i += gridDim.x * blockDim.x)
        o4[i] = z;
    if (blockIdx.x == 0 && threadIdx.x < 128)  // sums[64] + sumsq[64]
        ws[threadIdx.x] = 0.f;
}

// ---------------------------------------------------------------------------
// Kernel 2: gather -> WMMA fp32 GEMM -> scatter-add
//   grid = (256, 27); block = 256 threads = 8 waves.
//   wave & 3  -> N-tile (16 output channels), wave >> 2 -> pair-tile slot.
//   Per 16-pair tile: A = gathered x rows (16x64), B = weight slice (64x16),
//   K=64 reduced as 16 chained V_WMMA_F32_16X16X4_F32 ops.
// ---------------------------------------------------------------------------
__global__ void __launch_bounds__(256)
sparse_conv_wmma(const float* __restrict__ x,
                 const float* __restrict__ w,
                 const int*   __restrict__ kin,
                 const int*   __restrict__ kout,
                 float*       __restrict__ y) {
    const int k    = blockIdx.y;
    const int wave = threadIdx.x >> 5;
    const int lane = threadIdx.x & 31;
    const int n0   = (wave & 3) << 4;   // output-channel tile base
    const int slot = wave >> 2;         // 0..1 pair-tile slot within block
    const int half = lane >> 4;         // lane half selects K pairs
    const int lm   = lane & 15;

    // ---- preload B fragments for this (k, n-tile): kept in registers ----
    // B is 4x16 (KxN) per chunk: VGPR0 holds K=0 (lanes 0-15) / K=2 (16-31),
    // VGPR1 holds K=1 / K=3.
    const float* Wk = w + k * (INC * OUTC);
    v2f b[16];
#pragma unroll
    for (int c = 0; c < 16; ++c) {
        const int i = c * 4 + 2 * half;
        b[c].x = Wk[i * OUTC + n0 + lm];
        b[c].y = Wk[(i + 1) * OUTC + n0 + lm];
    }

    const int* kin_k  = kin  + k * N_PAIRS;
    const int* kout_k = kout + k * N_PAIRS;

    for (int t = blockIdx.x * 2 + slot; t < TILES; t += gridDim.x * 2) {
        const int base  = t * 16;
        const int inRow = kin_k[base + lm];          // lane half duplicates; fine
        const float* xr = x + (long)inRow * INC;

        v8f acc = {};
#pragma unroll
        for (int c = 0; c < 16; ++c) {
            // A is 16x4 (MxK): lane = M row, VGPR0 = K{0,2}, VGPR1 = K{1,3}
            v2f a;
            a.x = xr[c * 4 + 2 * half];
            a.y = xr[c * 4 + 2 * half + 1];
            acc = __builtin_amdgcn_wmma_f32_16x16x4_f32(
                      /*neg_a=*/false, a, /*neg_b=*/false, b[c],
                      /*c_mod=*/(short)0, acc,
                      /*reuse_a=*/false, /*reuse_b=*/false);
        }

        // D layout 16x16 f32: VGPR r -> M=r (lanes 0-15), M=r+8 (lanes 16-31)
#pragma unroll
        for (int r = 0; r < 8; ++r) {
            const int m    = r + half * 8;
            const int oRow = kout_k[base + m];
            atomicAddF32(&y[(long)oRow * OUTC + n0 + lm], acc[r]);
        }
    }
}

// ---------------------------------------------------------------------------
// Kernel 3: per-channel sum / sum-of-squares over all N_VOX rows
// ---------------------------------------------------------------------------
__global__ void __launch_bounds__(256)
bn_stats_kernel(const float* __restrict__ y, float* __restrict__ sums) {
    __shared__ float ssum[256];
    __shared__ float ssq[256];
    const int c    = threadIdx.x & 63;   // channel
    const int rgrp = threadIdx.x >> 6;   // 0..3 row groups per block
    float s = 0.f, q = 0.f;
    for (int r = blockIdx.x * 4 + rgrp; r < N_VOX; r += gridDim.x * 4) {
        const float v = y[(long)r * OUTC + c];
        s += v;
        q += v * v;
    }
    ssum[threadIdx.x] = s;
    ssq[threadIdx.x]  = q;
    __syncthreads();
    if (threadIdx.x < 64) {
        s = ssum[c] + ssum[c + 64] + ssum[c + 128] + ssum[c + 192];
        q = ssq[c]  + ssq[c + 64]  + ssq[c + 128]  + ssq[c + 192];
        atomicAddF32(&sums[c], s);
        atomicAddF32(&sums[64 + c], q);
    }
}

// ---------------------------------------------------------------------------
// Kernel 4: in-place BatchNorm (training stats, biased var) + ReLU
// ---------------------------------------------------------------------------
__global__ void __launch_bounds__(256)
bn_relu_kernel(float* __restrict__ y,
               const float* __restrict__ sums,
               const float* __restrict__ gamma,
               const float* __restrict__ beta) {
    const long n      = (long)N_VOX * OUTC;
    const long idx0   = (long)blockIdx.x * blockDim.x + threadIdx.x;
    const long stride = (long)gridDim.x * blockDim.x;   // multiple of 64
    const int  c      = (int)(idx0 & 63);               // constant per thread
    const float inv_n = 1.0f / (float)N_VOX;
    const float mean  = sums[c] * inv_n;
    const float var   = sums[64 + c] * inv_n - mean * mean;
    const float sc    = rsqrtf(var + BN_EPS) * gamma[c];
    const float sh    = beta[c] - mean * sc;
    for (long i = idx0; i < n; i += stride) {
        const float v = y[i] * sc + sh;
        y[i] = fmaxf(v, 0.f);
    }
}

// ---------------------------------------------------------------------------
extern "C" void kernel_launch(void* const* d_in, const int* in_sizes, int n_in,
                              void* d_out, int out_size, void* d_ws, size_t ws_size,
                              hipStream_t stream) {
    const float* x     = (const float*)d_in[0];
    const float* w     = (const float*)d_in[1];
    const float* gamma = (const float*)d_in[2];
    const float* beta  = (const float*)d_in[3];
    const int*   kin   = (const int*)d_in[4];
    const int*   kout  = (const int*)d_in[5];
    float*       out   = (float*)d_out;
    float*       ws    = (float*)d_ws;

    (void)in_sizes; (void)n_in; (void)out_size; (void)ws_size;

    // 1) zero accumulator + stats scratch
    zero_kernel<<<2048, 256, 0, stream>>>(out, ws, (N_VOX * OUTC) / 4);

    // 2) gather-WMMA-scatter sparse conv
    sparse_conv_wmma<<<dim3(256, KOFF), 256, 0, stream>>>(x, w, kin, kout, out);

    // 3) batch-norm statistics
    bn_stats_kernel<<<512, 256, 0, stream>>>(out, ws);

    // 4) normalize + ReLU in place
    bn_relu_kernel<<<1024, 256, 0, stream>>>(out, ws, gamma, beta);
}